// QuantumLayer_42580305772775
// MI455X (gfx1250) — compile-verified
//
#include <hip/hip_runtime.h>
#include <math.h>

// ---------------------------------------------------------------------------
// Quantum layer: 14 qubits, batch 256.
//   Kernel 1 (qsim):   one workgroup per sample; full statevector (16384
//                      complex = 128KB) lives in LDS. Encoding + RX layer 0
//                      folded into a per-qubit product state (register-cached
//                      2-vectors); both CNOT rings folded into GF(2) index
//                      permutations; only the 14 layer-1 RX gates sweep the
//                      LDS state. Writes |psi|^2 (f32) to workspace.
//   Kernel 2 (wmma):   OUT[256,14] = PROBS[256,16384] x SIGN[16384,14]
//                      via V_WMMA_F32_16X16X4_F32 (f32 A/B -> exact signs,
//                      f32 accumulate). SIGN (+/-1) synthesized with sign-bit
//                      tricks (3 VALU/element), A loaded as b64.
// ---------------------------------------------------------------------------

#define NQ     14
#define NSTATE 16384         // 1 << NQ
#define BLK    256

typedef float v2f __attribute__((ext_vector_type(2)));
typedef float v8f __attribute__((ext_vector_type(8)));

// Composed read-index map for the ring CNOT(0,1),CNOT(1,2),...,CNOT(13,0):
//   state_after[b] = state_before[g(b)],  g = f1 o f2 o ... o f14
// (f14 applied to b first).  Wire q sits at bit position (13 - q) (wire 0 = MSB).
__device__ __forceinline__ unsigned ring_perm(unsigned b) {
  unsigned m = b;
  // f14 = CNOT(ctrl=13, targ=0)
  m ^= (m & 1u) << 13;
  // f13..f1 = CNOT(c, c+1) for c = 12..0
#pragma unroll
  for (int c = 12; c >= 0; --c) {
    unsigned cb = (m >> (13 - c)) & 1u;
    m ^= cb << (13 - (c + 1));
  }
  return m;
}

__global__ __launch_bounds__(BLK)
void qsim_kernel(const float* __restrict__ x, const float* __restrict__ w,
                 float* __restrict__ probs_ws, float* __restrict__ out,
                 int use_probs) {
  extern __shared__ float smem[];
  float* sr  = smem;                 // [16384] real
  float* si  = smem + NSTATE;        // [16384] imag
  float* vre = smem + 2 * NSTATE;    // [28] per-qubit vector real
  float* vim = vre + 2 * NQ;         // [28] per-qubit vector imag

  const int b   = blockIdx.x;        // sample
  const int tid = threadIdx.x;

  // ---- Stage A: fold angle-encoding (RZ@RY column 0) + layer-0 RX ---------
  if (tid < NQ) {
    float th = tanhf(x[b * NQ + tid]) * 3.14159265358979323846f;
    float h  = 0.5f * th;
    float s, c;
    sincosf(h, &s, &c);
    // |psi_q> = gate * |0> = [g00, g10]:  g00 = e^{-i h} c,  g10 = e^{+i h} s
    float g00r = c * c, g00i = -c * s;
    float g10r = c * s, g10i = s * s;
    float ws0, wc0;
    sincosf(0.5f * w[tid], &ws0, &wc0);            // weights[0][q]
    // v = RX(w0) * [g00, g10]
    vre[2 * tid + 0] = wc0 * g00r + ws0 * g10i;
    vim[2 * tid + 0] = wc0 * g00i - ws0 * g10r;
    vre[2 * tid + 1] = ws0 * g00i + wc0 * g10r;
    vim[2 * tid + 1] = -ws0 * g00r + wc0 * g10i;
  }
  __syncthreads();

  // Register-cache the 14 per-qubit 2-vectors (loop-invariant for Stage B).
  float vr[2 * NQ], vi[2 * NQ];
#pragma unroll
  for (int j = 0; j < 2 * NQ; ++j) { vr[j] = vre[j]; vi[j] = vim[j]; }

  // ---- Stage B: product state with CNOT-ring #1 folded into the index ----
  for (int k = tid; k < NSTATE; k += BLK) {
    unsigned m = ring_perm((unsigned)k);
    float ar = 1.f, ai = 0.f;
#pragma unroll
    for (int q = 0; q < NQ; ++q) {
      unsigned bit = (m >> (13 - q)) & 1u;
      float br = bit ? vr[2 * q + 1] : vr[2 * q];   // v_cndmask, no LDS read
      float bi = bit ? vi[2 * q + 1] : vi[2 * q];
      float nr = ar * br - ai * bi;
      float ni = ar * bi + ai * br;
      ar = nr; ai = ni;
    }
    sr[k] = ar;
    si[k] = ai;
  }
  __syncthreads();

  // ---- Stage C: layer-1 RX gates, in-place pair updates in LDS ------------
#pragma unroll 1
  for (int q = 0; q < NQ; ++q) {
    float ws1, wc1;
    sincosf(0.5f * w[NQ + q], &ws1, &wc1);         // weights[1][q]
    const int p = 13 - q;
    const unsigned lowmask = (1u << p) - 1u;
    for (int t = tid; t < NSTATE / 2; t += BLK) {
      unsigned lo = (unsigned)t & lowmask;
      unsigned hi = ((unsigned)t >> p) << (p + 1);
      unsigned k0 = hi | lo;
      unsigned k1 = k0 | (1u << p);
      float r0 = sr[k0], i0 = si[k0];
      float r1 = sr[k1], i1 = si[k1];
      sr[k0] =  wc1 * r0 + ws1 * i1;
      si[k0] =  wc1 * i0 - ws1 * r1;
      sr[k1] =  ws1 * i0 + wc1 * r1;
      si[k1] = -ws1 * r0 + wc1 * i1;
    }
    __syncthreads();
  }

  // ---- Stage D/E: CNOT-ring #2 folded into the readout gather -------------
  if (use_probs) {
    float* pw = probs_ws + (size_t)b * NSTATE;
    for (int k = tid; k < NSTATE; k += BLK) {
      unsigned m = ring_perm((unsigned)k);
      float r = sr[m], im = si[m];
      pw[k] = r * r + im * im;                     // coalesced global write
    }
  } else {
    // Fallback (workspace too small): in-block reduction to out directly.
    float acc[NQ];
#pragma unroll
    for (int q = 0; q < NQ; ++q) acc[q] = 0.f;
    for (int k = tid; k < NSTATE; k += BLK) {
      unsigned m = ring_perm((unsigned)k);
      float r = sr[m], im = si[m];
      float p2 = r * r + im * im;
#pragma unroll
      for (int q = 0; q < NQ; ++q)
        acc[q] += ((k >> (13 - q)) & 1u) ? -p2 : p2;
    }
    __syncthreads();
#pragma unroll
    for (int q = 0; q < NQ; ++q) sr[tid * NQ + q] = acc[q];
    __syncthreads();
    for (int stride = BLK / 2; stride > 0; stride >>= 1) {
      if (tid < stride) {
#pragma unroll
        for (int q = 0; q < NQ; ++q)
          sr[tid * NQ + q] += sr[(tid + stride) * NQ + q];
      }
      __syncthreads();
    }
    if (tid < NQ) out[b * NQ + tid] = sr[tid];
  }
}

// ---------------------------------------------------------------------------
// OUT[256,14] = PROBS[256,16384] x SIGN[16384,14], one wave per 16-sample tile.
// A fragment (16x4 f32): lane L holds M = L%16, VGPR j holds K = 2*(L/16)+j.
// B fragment (4x16 f32): lane L holds N = L%16, VGPR j holds K = 2*(L/16)+j.
// D (16x16 f32): VGPR r, lane L -> (M = r + 8*(L/16), N = L%16).
// SIGN[k][n] = +/-1 built from bit (13-n) of k via sign-bit ops (no compares).
// ---------------------------------------------------------------------------
__global__ __launch_bounds__(32)
void expval_wmma_kernel(const float* __restrict__ probs, float* __restrict__ out) {
  const int tile  = blockIdx.x;        // samples [tile*16, tile*16+16)
  const int lane  = threadIdx.x;       // 0..31, EXEC all ones (WMMA requirement)
  const int m     = lane & 15;         // A row within tile == B column
  const int khalf = lane >> 4;         // 0 or 1
  const bool colok = (m < NQ);

  const float* arow  = probs + (size_t)(tile * 16 + m) * NSTATE;
  const v2f*   arow2 = (const v2f*)arow;

  // Per-lane constants for sign synthesis: +/-1.0f = 0x3f800000 | signbit.
  const int      bitpos   = colok ? (13 - m) : 0;
  const unsigned shift    = (unsigned)(31 - bitpos);
  const unsigned onebits  = colok ? 0x3f800000u : 0u;   // 0 -> column is 0.0f
  const unsigned signmask = colok ? 0x80000000u : 0u;

  v8f acc = {0.f, 0.f, 0.f, 0.f, 0.f, 0.f, 0.f, 0.f};

#pragma unroll 8
  for (int k0 = 0; k0 < NSTATE; k0 += 4) {
    const int ka = k0 + 2 * khalf;
    if ((k0 & 511) == 0)
      __builtin_prefetch(arow + k0 + 1024, 0, 1);  // global_prefetch_b8

    // One global_load_b64 for the two consecutive A elements of this lane.
    v2f a = arow2[(k0 >> 1) + khalf];

    v2f bb;
    bb.x = __uint_as_float(onebits | (((unsigned)ka       << shift) & signmask));
    bb.y = __uint_as_float(onebits | (((unsigned)(ka + 1) << shift) & signmask));

    // D = A * B + C   (v_wmma_f32_16x16x4_f32)
    acc = __builtin_amdgcn_wmma_f32_16x16x4_f32(
        /*neg_a=*/false, a, /*neg_b=*/false, bb,
        /*c_mod=*/(short)0, acc, /*reuse_a=*/false, /*reuse_b=*/false);
  }

#pragma unroll
  for (int r = 0; r < 8; ++r) {
    int M = r + 8 * khalf;
    if (colok) out[(size_t)(tile * 16 + M) * NQ + m] = acc[r];
  }
}

extern "C" void kernel_launch(void* const* d_in, const int* in_sizes, int n_in,
                              void* d_out, int out_size, void* d_ws, size_t ws_size,
                              hipStream_t stream) {
  const float* x = (const float*)d_in[0];   // [256,14] f32
  const float* w = (const float*)d_in[1];   // [2,14]  f32
  float* out = (float*)d_out;               // [256,14] f32

  const int batch = in_sizes[0] / NQ;       // 256
  const size_t probs_bytes = (size_t)batch * NSTATE * sizeof(float);  // 16 MB
  const int use_probs = (ws_size >= probs_bytes && (batch % 16) == 0) ? 1 : 0;
  float* probs = (float*)d_ws;

  const size_t shmem = (2 * NSTATE + 4 * NQ) * sizeof(float);  // ~128.2 KB LDS

  qsim_kernel<<<batch, BLK, shmem, stream>>>(x, w, probs, out, use_probs);
  if (use_probs)
    expval_wmma_kernel<<<batch / 16, 32, 0, stream>>>(probs, out);
}